// Attention2D_70660801954139
// MI455X (gfx1250) — compile-verified
//
#include <hip/hip_runtime.h>
#include <math.h>

// CDNA5 / gfx1250 fused Attention2D kernel, v2.
// Memory-bound (~430 MB -> ~18.5us at 23.3 TB/s). All dense matmuls on
// V_WMMA_F32_16X16X4_F32 (f32 fidelity; HBM traffic is f32 either way).
// v2 minimizes per-group issue slots: weight B-fragments live in registers,
// A-fragments gathered straight from global, C tiles stay in registers with
// shfl_xor(16) column reductions; LDS only where a real C->A transpose is
// needed (U tile, hidden tiles) plus batched QP / X tiles. One wave owns a
// batch of 16 (r,s) groups; qp and x@oW are batched 16-row WMMA GEMMs.

typedef float v2f __attribute__((ext_vector_type(2)));
typedef float v8f __attribute__((ext_vector_type(8)));

static constexpr int kN = 10;           // n_src
static constexpr int kD = 32;           // feature dim
static constexpr int kNB = 16;          // groups per wave (batch = one WMMA tile of rows)
static constexpr float kTiny = 1e-6f;
static constexpr int WAVES_PER_BLOCK = 4;

// per-wave LDS arena (floats)
static constexpr int A_QP = 0;      // 16x32 batched qp rows
static constexpr int A_XM = 512;    // 16x32 batched x rows
static constexpr int A_U  = 1024;   // 16x32 u tile (C->A transpose staging)
static constexpr int A_H  = 1536;   // 16x16 hidden tile staging
static constexpr int ARENA = 1792;

#define V8F0 v8f{0.f,0.f,0.f,0.f,0.f,0.f,0.f,0.f}

__device__ __forceinline__ v2f ld2(const float* p) {
  const float2 t = *(const float2*)p;
  v2f r; r.x = t.x; r.y = t.y; return r;
}
// A fragment (16x4 f32) from an LDS tile: lanes 0-15 row=lane K={4j,4j+1},
// lanes 16-31 row=lane-16 K={4j+2,4j+3}.
__device__ __forceinline__ v2f frag_a_lds(const float* T, int stride, int kstep, int lane) {
  return ld2(T + (lane & 15) * stride + kstep * 4 + 2 * (lane >> 4));
}
// B fragment (4x16 f32) from a row-major KxN matrix in global memory.
__device__ __forceinline__ v2f frag_b_g(const float* W, int ncols, int kstep, int ctile, int lane) {
  const int col = (lane & 15) + ctile * 16;
  const int r0  = kstep * 4 + 2 * (lane >> 4);
  v2f b; b.x = W[r0 * ncols + col]; b.y = W[(r0 + 1) * ncols + col]; return b;
}
// B fragment from a Kx4 matrix, zero-padded to 16 columns.
__device__ __forceinline__ v2f frag_b_g4(const float* W, int kstep, int lane) {
  const int col = lane & 15;
  const int r0  = kstep * 4 + 2 * (lane >> 4);
  v2f b; b.x = 0.f; b.y = 0.f;
  if (col < 4) { b.x = W[r0 * 4 + col]; b.y = W[(r0 + 1) * 4 + col]; }
  return b;
}
__device__ __forceinline__ v8f wmma_step(v2f a, v2f b, v8f c) {
  return __builtin_amdgcn_wmma_f32_16x16x4_f32(false, a, false, b, (short)0, c, false, false);
}

__global__ __launch_bounds__(WAVES_PER_BLOCK * 32)
void attention2d_kernel(const float* __restrict__ q, const float* __restrict__ k,
                        const float* __restrict__ pos, const int* __restrict__ mask,
                        const float* __restrict__ Wq, const float* __restrict__ Wk,
                        const float* __restrict__ Wv,
                        const float* __restrict__ pW1, const float* __restrict__ pb1,
                        const float* __restrict__ pW2, const float* __restrict__ pb2,
                        const float* __restrict__ aW1, const float* __restrict__ ab1,
                        const float* __restrict__ aW2, const float* __restrict__ ab2,
                        const float* __restrict__ oW, const float* __restrict__ ob,
                        float* __restrict__ out_x, float* __restrict__ out_attn,
                        float* __restrict__ out_kstd, float* __restrict__ out_nks,
                        int groups) {
  __shared__ float lds[WAVES_PER_BLOCK * ARENA];
  const int tid   = threadIdx.x;
  const int lane  = tid & 31;
  const int wave  = tid >> 5;
  const int col16 = lane & 15;
  const int half  = lane >> 4;
  const bool hicl = (half != 0);

  float* AR  = lds + wave * ARENA;   // wave-private: no cross-wave sharing, no barriers
  float* QP  = AR + A_QP;
  float* XM  = AR + A_XM;
  float* UST = AR + A_U;
  float* HST = AR + A_H;

  const int wgid   = blockIdx.x * WAVES_PER_BLOCK + wave;
  const int stride = gridDim.x * WAVES_PER_BLOCK;

  // ---- loop-invariant weight B-fragments -> registers (read once via L2) ----
  v2f WkB[2][8], WvB[2][8], aW1B[8], aW2B[2], pW2B[2], pW1B;
#pragma unroll
  for (int t = 0; t < 2; ++t)
#pragma unroll
    for (int j = 0; j < 8; ++j) {
      WkB[t][j] = frag_b_g(Wk, 32, j, t, lane);
      WvB[t][j] = frag_b_g(Wv, 32, j, t, lane);
    }
#pragma unroll
  for (int j = 0; j < 8; ++j) aW1B[j] = frag_b_g4(aW1, j, lane);
#pragma unroll
  for (int t = 0; t < 2; ++t) { aW2B[t] = frag_b_g(aW2, 32, 0, t, lane); pW2B[t] = frag_b_g(pW2, 32, 0, t, lane); }
  pW1B = frag_b_g4(pW1, 0, lane);

  // bias scalars (per-lane columns)
  const float pb1v = (col16 < 4) ? pb1[col16] : 0.f;
  const float ab1v = (col16 < 4) ? ab1[col16] : 0.f;
  float pb2v[2] = { pb2[col16], pb2[col16 + 16] };
  float ab2v[2] = { ab2[col16], ab2[col16 + 16] };
  float obv [2] = { ob [col16], ob [col16 + 16] };

  // ---- batched qp = q_rows(16 groups) @ Wq (WMMA) -> QP tile in LDS ----
  {
    int gq = wgid + col16 * stride;
    if (gq > groups - 1) gq = groups - 1;
    const float* qrow = q + (size_t)gq * kD + 2 * half;
    v2f qA[8];
#pragma unroll
    for (int j = 0; j < 8; ++j) qA[j] = ld2(qrow + j * 4);
#pragma unroll
    for (int t = 0; t < 2; ++t) {
      v8f acc = V8F0;
#pragma unroll
      for (int j = 0; j < 8; ++j) acc = wmma_step(qA[j], frag_b_g(Wq, 32, j, t, lane), acc);
#pragma unroll
      for (int v = 0; v < 8; ++v) QP[(v + 8 * half) * 32 + col16 + 16 * t] = acc[v];
    }
  }

  // ---- per-group pipeline (16 groups per wave) ----
#pragma unroll 1
  for (int i = 0; i < kNB; ++i) {
    const int g = wgid + i * stride;
    const bool valid = (g < groups);
    const int gl = valid ? g : (groups - 1);

    // A fragments straight from global (row clamped to 9 for the pad rows:
    // finite duplicate data, row-quarantined by WMMA semantics).
    const int rowc = (col16 < kN) ? col16 : (kN - 1);
    const float* kbase = k + (size_t)gl * kN * kD + rowc * kD + 2 * half;
    v2f kA[8];
#pragma unroll
    for (int j = 0; j < 8; ++j) kA[j] = ld2(kbase + j * 4);
    const v2f posA = ld2(pos + (size_t)gl * kN * 4 + rowc * 4 + 2 * half);

    // mask -> per-row weights (uniform across lanes; 5 x b64 broadcast loads)
    float wn[kN]; float cnt = 0.f;
    const int2* mp = (const int2*)(mask + (size_t)gl * kN);
#pragma unroll
    for (int p = 0; p < 5; ++p) {
      const int2 mm = mp[p];
      wn[2*p]   = (mm.x != 0) ? 1.f : 0.f;
      wn[2*p+1] = (mm.y != 0) ? 1.f : 0.f;
      cnt += wn[2*p] + wn[2*p+1];
    }
    const bool all_inv = (cnt < 0.5f);
    const float cnt_eff = all_inv ? (float)kN : cnt;
    if (all_inv) {
#pragma unroll
      for (int n = 0; n < kN; ++n) wn[n] = 1.f;
    }
    // row weight seen by this lane's C-fragment slots (lo half: rows v, hi: rows 8+v)
    float rw[8];
#pragma unroll
    for (int v = 0; v < 8; ++v) rw[v] = hicl ? ((v < 2) ? wn[8 + v] : 0.f) : wn[v];

    // ---- kp = k@Wk, vp = k@Wv : pure register WMMA ----
    v8f kp[2], vp[2];
#pragma unroll
    for (int t = 0; t < 2; ++t) {
      v8f acc = V8F0;
#pragma unroll
      for (int j = 0; j < 8; ++j) acc = wmma_step(kA[j], WkB[t][j], acc);
      kp[t] = acc;
    }
#pragma unroll
    for (int t = 0; t < 2; ++t) {
      v8f acc = V8F0;
#pragma unroll
      for (int j = 0; j < 8; ++j) acc = wmma_step(kA[j], WvB[t][j], acc);
      vp[t] = acc;
    }

    // ---- pemb = relu(pos@pW1 + pb1) @ pW2 + pb2 ----
    v8f pe[2];
    {
      v8f h = wmma_step(posA, pW1B, V8F0);
#pragma unroll
      for (int v = 0; v < 8; ++v) HST[(v + 8 * half) * 16 + col16] = fmaxf(h[v] + pb1v, 0.f);
      const v2f hA = frag_a_lds(HST, 16, 0, lane);
#pragma unroll
      for (int t = 0; t < 2; ++t) {
        v8f p_ = wmma_step(hA, pW2B[t], V8F0);
#pragma unroll
        for (int v = 0; v < 8; ++v) p_[v] += pb2v[t];
        pe[t] = p_;
      }
    }

    // ---- masked stats over rows: lane-partial sums + shfl_xor(16) combine ----
    float kstdv[2], nksv[2];
#pragma unroll
    for (int t = 0; t < 2; ++t) {
      float s1 = 0.f, sa = 0.f;
#pragma unroll
      for (int v = 0; v < 8; ++v) { s1 = fmaf(rw[v], kp[t][v], s1); sa = fmaf(rw[v], fabsf(kp[t][v]), sa); }
      s1 += __shfl_xor(s1, 16, 32);
      sa += __shfl_xor(sa, 16, 32);
      const float mean = s1 / cnt_eff, mabs = sa / cnt_eff;
      float sv = 0.f;
#pragma unroll
      for (int v = 0; v < 8; ++v) { const float d = kp[t][v] - mean; sv = fmaf(rw[v], d * d, sv); }
      sv += __shfl_xor(sv, 16, 32);
      const float var = sv / fmaxf(cnt_eff - 1.f, 1.f);
      const bool single = (cnt_eff < 1.5f);
      const float sd = sqrtf(single ? 1.f : var);
      kstdv[t] = single ? 0.f : sd;
      nksv[t]  = single ? 0.f : sd / (mabs + kTiny);
    }
    if (valid) {
      out_kstd[(size_t)g * kD + lane] = hicl ? kstdv[1] : kstdv[0];
      out_nks [(size_t)g * kD + lane] = hicl ? nksv[1]  : nksv[0];
    }

    // ---- u = kp - qp + pemb -> LDS (the one true C->A transpose) ----
#pragma unroll
    for (int t = 0; t < 2; ++t) {
      const float qpv = QP[i * 32 + col16 + 16 * t];
#pragma unroll
      for (int v = 0; v < 8; ++v) UST[(v + 8 * half) * 32 + col16 + 16 * t] = kp[t][v] + pe[t][v] - qpv;
    }

    // ---- scores = relu(u@aW1 + ab1) @ aW2 + ab2 ----
    v8f sc[2];
    {
      v8f hs = V8F0;
#pragma unroll
      for (int j = 0; j < 8; ++j) hs = wmma_step(frag_a_lds(UST, 32, j, lane), aW1B[j], hs);
#pragma unroll
      for (int v = 0; v < 8; ++v) HST[(v + 8 * half) * 16 + col16] = fmaxf(hs[v] + ab1v, 0.f);
      const v2f hA = frag_a_lds(HST, 16, 0, lane);
#pragma unroll
      for (int t = 0; t < 2; ++t) {
        v8f s_ = wmma_step(hA, aW2B[t], V8F0);
#pragma unroll
        for (int v = 0; v < 8; ++v) s_[v] += ab2v[t];
        sc[t] = s_;
      }
    }

    // ---- masked softmax over rows + attn store + weighted sum ----
    float xp[2];
#pragma unroll
    for (int t = 0; t < 2; ++t) {
      float mx = -3.0e38f;
#pragma unroll
      for (int v = 0; v < 8; ++v) mx = fmaxf(mx, (rw[v] > 0.5f) ? sc[t][v] : -3.0e38f);
      mx = fmaxf(mx, __shfl_xor(mx, 16, 32));
      float es = 0.f, at[8];
#pragma unroll
      for (int v = 0; v < 8; ++v) { at[v] = (rw[v] > 0.5f) ? __expf(sc[t][v] - mx) : 0.f; es += at[v]; }
      es += __shfl_xor(es, 16, 32);
      const float inv = 1.f / es;
      float xa = 0.f;
#pragma unroll
      for (int v = 0; v < 8; ++v) {
        const float a = at[v] * inv;
        const int row = v + 8 * half;
        if (valid && (!hicl || v < 2))
          out_attn[((size_t)g * kN + row) * kD + col16 + 16 * t] = a;
        xa = fmaf(vp[t][v] + pe[t][v], a, xa);
      }
      xa += __shfl_xor(xa, 16, 32);
      xp[t] = xa;
    }
    XM[i * 32 + lane] = hicl ? xp[1] : xp[0];   // buffer x row for the batched oW GEMM
  }

  // ---- batched out = X(16 rows) @ oW + ob (WMMA) ----
  {
    v2f xA[8];
#pragma unroll
    for (int j = 0; j < 8; ++j) xA[j] = frag_a_lds(XM, 32, j, lane);
#pragma unroll
    for (int t = 0; t < 2; ++t) {
      v8f acc = V8F0;
#pragma unroll
      for (int j = 0; j < 8; ++j) acc = wmma_step(xA[j], frag_b_g(oW, 32, j, t, lane), acc);
#pragma unroll
      for (int v = 0; v < 8; ++v) {
        const int row = v + 8 * half;          // batch slot
        const int gb = wgid + row * stride;
        if (gb < groups) out_x[(size_t)gb * kD + col16 + 16 * t] = acc[v] + obv[t];
      }
    }
  }
}

extern "C" void kernel_launch(void* const* d_in, const int* in_sizes, int n_in,
                              void* d_out, int out_size, void* d_ws, size_t ws_size,
                              hipStream_t stream) {
  const float* q    = (const float*)d_in[0];
  const float* k    = (const float*)d_in[1];
  const float* pos  = (const float*)d_in[2];
  const int*   mask = (const int*)  d_in[3];
  const float* Wq   = (const float*)d_in[4];
  const float* Wk   = (const float*)d_in[5];
  const float* Wv   = (const float*)d_in[6];
  const float* pW1  = (const float*)d_in[7];
  const float* pb1  = (const float*)d_in[8];
  const float* pW2  = (const float*)d_in[9];
  const float* pb2  = (const float*)d_in[10];
  const float* aW1  = (const float*)d_in[11];
  const float* ab1  = (const float*)d_in[12];
  const float* aW2  = (const float*)d_in[13];
  const float* ab2  = (const float*)d_in[14];
  const float* oW   = (const float*)d_in[15];
  const float* ob   = (const float*)d_in[16];

  const int groups = in_sizes[0] / kD;  // R*S
  float* out       = (float*)d_out;
  float* out_x     = out;
  float* out_attn  = out + (size_t)groups * kD;
  float* out_kstd  = out_attn + (size_t)groups * kN * kD;
  float* out_nks   = out_kstd + (size_t)groups * kD;

  const int per_block = kNB * WAVES_PER_BLOCK;                 // groups per block
  const int blocks = (groups + per_block - 1) / per_block;     // 2048 for R*S=131072
  dim3 grid(blocks), block(WAVES_PER_BLOCK * 32);
  attention2d_kernel<<<grid, block, 0, stream>>>(
      q, k, pos, mask, Wq, Wk, Wv, pW1, pb1, pW2, pb2,
      aW1, ab1, aW2, ab2, oW, ob,
      out_x, out_attn, out_kstd, out_nks, groups);
}